// PseudoMGLSADigitalFilter_74741020885330
// MI455X (gfx1250) — compile-verified
//
#include <hip/hip_runtime.h>

// PseudoMGLSADigitalFilter (alpha=0, gamma=0, frame_period=1, minimum phase)
// y[t] = x[t] + sum_{n=1..30} u_n[t],  u_n = (1/n) * TV-FIR25(u_{n-1}), u_0 = x
//
// Single fused kernel: each block owns CHUNK samples of one batch row with a
// 720-sample recomputed left halo (24 samples/iter x 30 iters), so all 30
// Taylor iterations run block-locally in LDS. mc (38MB) stays resident in the
// 192MB L2 across iterations -> L2-bandwidth bound, one launch.
//
// Inner 25-tap time-varying dot product rides the CDNA5 matrix path: 7 chained
// V_WMMA_F32_16X16X4_F32 (K padded 25->28); the result diagonal holds the 16
// output samples. Everything is branchless:
//  - u tile stored TIME-REVERSED in LDS so each B fragment pair comes out of a
//    single ds_load_2addr_b32 already in {x,y} register order (no swap movs)
//  - negative lags read the zero end-pad of the reversed tile
//  - mc row index clamped instead of masked (halo stays exactly 0 by induction
//    from the causal zero pad; tail rows are finite and never stored)
//  - diagonal store is full-EXEC: lanes 8..23 write a dump word in the pad
//    (index REGION+31; B reads never go past REGION+26)
//  - y accumulator lives in 4 VGPRs per thread (CHUNK/NTHREADS == 4)
//  - one barrier per Taylor iteration: after it, y-accum only READS the new
//    buffer while the next iteration WRITES the other buffer (no conflict).

typedef __attribute__((ext_vector_type(2))) float v2f;
typedef __attribute__((ext_vector_type(8))) float v8f;

#define T_LEN    48000
#define M1       25                 // taps = M+1
#define NITER    30                 // Taylor order
#define CHUNK    1024               // output samples per block
#define HALO     720                // 24 * NITER dependency halo
#define REGION   (CHUNK + HALO)     // 1744, multiple of 16
#define NGROUPS  (REGION / 16)      // 109 wmma row-groups per pass
#define NTHREADS 256                // 8 waves (wave32)
#define NWAVES   (NTHREADS / 32)
#define YPT      (CHUNK / NTHREADS) // 4 y samples per thread, in VGPRs
#define PADL     32                 // zero pad at END of reversed tile (>= 28)
#define USZ      (REGION + PADL)
#define DUMP     (USZ - 1)          // lanes 8..23 park stores here

// reversed tile: physical p holds logical sample l = REGION-1-p

__global__ __launch_bounds__(NTHREADS)
void mglsa_taylor_wmma_kernel(const float* __restrict__ x,
                              const float* __restrict__ mc,
                              float* __restrict__ y)
{
    __shared__ float uA[USZ];
    __shared__ float uB[USZ];

    const int b    = blockIdx.y;
    const int cs   = blockIdx.x * CHUNK;   // chunk start (global t)
    const int R    = cs - HALO;            // region start (may be < 0)
    const int tid  = threadIdx.x;
    const int lane = tid & 31;
    const int wave = tid >> 5;

    const float* __restrict__ xb  = x  + (size_t)b * T_LEN;
    const float* __restrict__ mcb = mc + (size_t)b * T_LEN * M1;

    // u_0 = x, reversed; pad (logical l<0) is zero. uB pad zeroed once.
    for (int p = tid; p < USZ; p += NTHREADS) {
        int l = (REGION - 1) - p;          // logical index (negative in pad)
        int t = R + l;
        uA[p] = (l >= 0 && (unsigned)t < (unsigned)T_LEN) ? xb[t] : 0.0f;
    }
    if (tid < PADL) uB[REGION + tid] = 0.0f;

    // y starts at x; kept in registers (thread owns samples tid + k*NTHREADS)
    float yreg[YPT];
#pragma unroll
    for (int k = 0; k < YPT; ++k) {
        int t = cs + tid + k * NTHREADS;
        yreg[k] = (t < T_LEN) ? xb[t] : 0.0f;
    }
    __syncthreads();

    float* ucur = uA;
    float* unxt = uB;

    const int  m       = lane & 15;          // row index within 16-group
    const int  half2   = (lane >> 4) << 1;   // K-offset of lane half (0 or 2)
    const bool hasDiag = (lane < 8) || (lane >= 24);
    const int  dslot   = (lane < 8) ? lane : (lane - 16); // diag sample idx

    for (int n = 1; n <= NITER; ++n) {
        const float inv_a = 1.0f / (float)n;

        for (int g = wave; g < NGROUPS; g += NWAVES) {
            const int l0   = g << 4;                 // local base in region
            const int trow = R + l0 + m;             // global time of this row
            const int tcl  = trow < 0 ? 0 : (trow >= T_LEN ? (T_LEN - 1) : trow);
            const float* __restrict__ mrow = mcb + (size_t)tcl * M1;
            // physical base of this lane's lag-0 B element in reversed tile
            const int pb = (REGION - 1) - (l0 + m - half2);

            // prefetch this lane's mc row for the next pass of this wave
            {
                int tpre = trow + (NWAVES << 4);
                tpre = tpre < 0 ? 0 : (tpre >= T_LEN ? (T_LEN - 1) : tpre);
                __builtin_prefetch(mcb + (size_t)tpre * M1, 0, 3);
            }

            v8f acc = {0.f, 0.f, 0.f, 0.f, 0.f, 0.f, 0.f, 0.f};

#pragma unroll
            for (int k0 = 0; k0 < 24; k0 += 4) {     // chunks with taps < 25
                const int j0 = k0 + half2;           // <= 22, j0+1 <= 23
                v2f a;
                a.x = mrow[j0];
                a.y = mrow[j0 + 1];
                v2f bb;
                bb.x = ucur[pb + k0];                // logical lag k0
                bb.y = ucur[pb + k0 + 1];            // logical lag k0+1
                acc = __builtin_amdgcn_wmma_f32_16x16x4_f32(
                          false, a, false, bb, (short)0, acc, false, false);
            }
            {   // last chunk: taps 24..27, only j=24 is real
                v2f a;
                float a0 = mrow[24];
                a.x = (half2 == 0) ? a0 : 0.0f;      // j=26 -> 0
                a.y = 0.0f;                          // j=25/27 -> 0
                v2f bb;
                bb.x = ucur[pb + 24];
                bb.y = ucur[pb + 25];
                acc = __builtin_amdgcn_wmma_f32_16x16x4_f32(
                          false, a, false, bb, (short)0, acc, false, false);
            }

            // Diagonal of D: lane v holds D[v][v] in acc[v] (v<8),
            // lane 24+v holds D[v+8][v+8] in acc[v]. Full-EXEC store:
            // lanes 8..23 park their (meaningless) value in the dump word.
            float dv = acc[0];
#pragma unroll
            for (int v = 1; v < 8; ++v)
                dv = ((lane & 7) == v) ? acc[v] : dv;

            const int pst = hasDiag ? ((REGION - 1) - (l0 + dslot)) : DUMP;
            unxt[pst] = dv * inv_a;
        }
        __syncthreads();

        // y += u_n over the valid window: logical HALO+i -> physical CHUNK-1-i.
        // Reads only the freshly written buffer; the next iteration writes the
        // other buffer, so no second barrier is required.
#pragma unroll
        for (int k = 0; k < YPT; ++k)
            yreg[k] += unxt[(CHUNK - 1) - (tid + k * NTHREADS)];

        float* tmp = ucur; ucur = unxt; unxt = tmp;
    }

    float* __restrict__ yb = y + (size_t)b * T_LEN;
#pragma unroll
    for (int k = 0; k < YPT; ++k) {
        int t = cs + tid + k * NTHREADS;
        if (t < T_LEN) yb[t] = yreg[k];
    }
}

extern "C" void kernel_launch(void* const* d_in, const int* in_sizes, int n_in,
                              void* d_out, int out_size, void* d_ws, size_t ws_size,
                              hipStream_t stream) {
    (void)n_in; (void)out_size; (void)d_ws; (void)ws_size;
    const float* x  = (const float*)d_in[0];   // (B, T) float32
    const float* mc = (const float*)d_in[1];   // (B, T, 25) float32
    float* yout     = (float*)d_out;           // (B, T) float32

    const int B = in_sizes[0] / T_LEN;         // 8
    dim3 grid((T_LEN + CHUNK - 1) / CHUNK, B); // 47 x 8 = 376 blocks
    mglsa_taylor_wmma_kernel<<<grid, NTHREADS, 0, stream>>>(x, mc, yout);
}